// PositionalEncoding_15212774162710
// MI455X (gfx1250) — compile-verified
//
#include <hip/hip_runtime.h>
#include <math.h>

// PositionalEncoding: out[b,s,j] = x[b,s,j] + (j even ? sin : cos)(s * 10000^(-2j/E))
// x: [8, 4096, 1024] fp32.  Pure streaming op: 256 MiB HBM traffic -> ~11.5 us
// at 23.3 TB/s.  One thread owns one (s, j..j+3) float4 and loops the 8 batch
// planes, so the sin/cos work is amortized 8x and the kernel is HBM-bound.

typedef __attribute__((ext_vector_type(4))) float v4f;

namespace {
constexpr int kBatch = 8;
constexpr int kSeq   = 4096;
constexpr int kEmb   = 1024;
constexpr int kVecPerRow = kEmb / 4;                       // 256 threads per row
constexpr long long kPlaneVec = (long long)kSeq * kVecPerRow; // float4s per batch plane
// angle = s * 10000^(-2j/E) = s * exp2(j * c),  c = -2*log2(10000)/E
constexpr float kC = -2.0f * 13.287712379549449f / (float)kEmb;
}

__global__ __launch_bounds__(kVecPerRow)
void pe_add_kernel(const float* __restrict__ x, float* __restrict__ out) {
  const int s  = blockIdx.x;            // one sequence row per block
  const int t  = threadIdx.x;           // float4 index within the row
  const int j0 = t * 4;                 // first embed index handled by this thread

  const v4f* __restrict__ xin =
      (const v4f*)x + ((long long)s * kVecPerRow + t);
  v4f* __restrict__ xo =
      (v4f*)out + ((long long)s * kVecPerRow + t);

  // Issue all 8 batch-plane loads up front (8 outstanding LOADcnt entries per
  // thread) so the trig below overlaps the memory latency.  Non-temporal:
  // each byte is touched exactly once -> don't pollute L2.
  v4f v[kBatch];
#pragma unroll
  for (int b = 0; b < kBatch; ++b) {
    v[b] = __builtin_nontemporal_load(xin + (long long)b * kPlaneVec);
  }

  // PE values for the 4 embed lanes (even j -> sin, odd j -> cos).
  // exp2 is native v_exp_f32; sin/cos are TRANS ops that co-execute with the
  // in-flight loads.
  const float pos = (float)s;
  const float a0 = pos * __builtin_exp2f((float)(j0 + 0) * kC);
  const float a1 = pos * __builtin_exp2f((float)(j0 + 1) * kC);
  const float a2 = pos * __builtin_exp2f((float)(j0 + 2) * kC);
  const float a3 = pos * __builtin_exp2f((float)(j0 + 3) * kC);
  const float pe0 = sinf(a0);
  const float pe1 = cosf(a1);
  const float pe2 = sinf(a2);
  const float pe3 = cosf(a3);

#pragma unroll
  for (int b = 0; b < kBatch; ++b) {
    v4f r = v[b];
    r.x += pe0;
    r.y += pe1;
    r.z += pe2;
    r.w += pe3;
    __builtin_nontemporal_store(r, xo + (long long)b * kPlaneVec);
  }
}

extern "C" void kernel_launch(void* const* d_in, const int* in_sizes, int n_in,
                              void* d_out, int out_size, void* d_ws, size_t ws_size,
                              hipStream_t stream) {
  (void)in_sizes; (void)n_in; (void)out_size; (void)d_ws; (void)ws_size;
  const float* x = (const float*)d_in[0];
  float* out = (float*)d_out;
  // 4096 blocks x 256 threads (8 wave32 waves per block); each block = one s row.
  pe_add_kernel<<<dim3(kSeq), dim3(kVecPerRow), 0, stream>>>(x, out);
}